// GaussianBlur_20933670601040
// MI455X (gfx1250) — compile-verified
//
#include <hip/hip_runtime.h>

// Gaussian 5x5 depthwise blur, per-batch separable kernel, reflect padding.
// imgs: [64,3,512,512] f32, xform_params: [64,2] f32 -> out: [64,3,512,512] f32.
//
// MI455X / gfx1250 (wave32) strategy:
//  - one 16x16 output tile per wave
//  - 20x20 reflect-padded patch pulled straight into per-wave LDS with
//    GLOBAL_LOAD_ASYNC_TO_LDS_B32 (ASYNCcnt / s_wait_asynccnt)
//  - vertical + horizontal 1D convolutions as banded-Toeplitz matmuls on
//    V_WMMA_F32_16X16X4_F32 (10 + 5 chained K=4 WMMAs per tile)
//  - Toeplitz weights built branch-free with v_cndmask select chains
//    (vertical-A and horizontal-B share the same band index per (t,slot))
//  - same-wave LDS store->load ordering via s_wait_dscnt 0

typedef __attribute__((ext_vector_type(2))) float v2f;
typedef __attribute__((ext_vector_type(8))) float v8f;

#define KS 5
#define PAD 2
#define IMG_H 512
#define IMG_W 512
#define CH 3
#define BATCH 64
#define TILE 16
#define PATCH 20            // TILE + KS - 1
#define PSTRIDE 21          // sP row stride (odd -> bank decorrelation)
#define VSTRIDE 22          // sV row stride (even -> 8B-aligned b64 A loads)
#define TILES_PER_PLANE 1024    // (512/16)^2
#define WAVES_PER_BLOCK 8

__device__ __forceinline__ void wave_ds_sync() {
  asm volatile("s_wait_dscnt 0" ::: "memory");
}
__device__ __forceinline__ void wave_async_sync() {
  asm volatile("s_wait_asynccnt 0" ::: "memory");
}

__global__ __launch_bounds__(256) void gauss5x5_wmma(
    const float* __restrict__ imgs,
    const float* __restrict__ xf,
    float* __restrict__ out) {
  // per-wave LDS regions (8 waves/block -> ~24.7 KB of the 320 KB WGP pool)
  __shared__ float sP[WAVES_PER_BLOCK][PATCH * PSTRIDE];  // 20x20 input patch
  __shared__ float sV[WAVES_PER_BLOCK][TILE * VSTRIDE];   // 16x20 vertical result

  const int tid  = threadIdx.x & 31;
  const int wave = threadIdx.x >> 5;
  const int idx  = tid & 15;   // column (B/C/D) or row (A) index
  const int half = tid >> 4;   // K slots {0,1} vs {2,3}; D rows +0 vs +8

  const int gtile = blockIdx.x * WAVES_PER_BLOCK + wave;
  const int plane = gtile >> 10;
  const int trem  = gtile & 1023;
  const int ty    = trem >> 5;
  const int tx    = trem & 31;
  const int b     = plane / CH;
  const int c     = plane % CH;

  const int y0 = ty * TILE;
  const int x0 = tx * TILE;

  // ---- async fill of the 20x20 reflect-padded patch (memory -> LDS) ----
  const size_t planeOff = (size_t)(b * CH + c) * (IMG_H * IMG_W);
  const float* img = imgs + planeOff;
  {
    // low 32 bits of the generic pointer == wave-relative LDS byte address
    // (flat-LDS mapping takes the low 32 address bits)
    const unsigned ldsbase = (unsigned)(uintptr_t)&sP[wave][0];
    for (int p = tid; p < PATCH * PATCH; p += 32) {
      const int j = p / PATCH;
      const int i = p - j * PATCH;
      int gy = y0 - PAD + j;
      gy = (gy < 0) ? -gy : gy;
      gy = (gy > IMG_H - 1) ? (2 * (IMG_H - 1) - gy) : gy;
      int gx = x0 - PAD + i;
      gx = (gx < 0) ? -gx : gx;
      gx = (gx > IMG_W - 1) ? (2 * (IMG_W - 1) - gx) : gx;
      const float* gp = img + (gy * IMG_W + gx);
      const unsigned la = ldsbase + (unsigned)((j * PSTRIDE + i) * 4);
      asm volatile("global_load_async_to_lds_b32 %0, %1, off"
                   :: "v"(la), "v"(gp) : "memory");
    }
  }

  // ---- per-batch separable Gaussian taps (registers only) ----
  float kx[KS], ky[KS];
  {
    const float sx = xf[b * 2 + 0];
    const float sy = xf[b * 2 + 1];
    float sxs = 0.0f, sys = 0.0f;
#pragma unroll
    for (int k = 0; k < KS; ++k) {
      const float d  = (float)(k - PAD);
      const float qx = d / sx;
      const float qy = d / sy;
      kx[k] = __expf(-0.5f * qx * qx);
      ky[k] = __expf(-0.5f * qy * qy);
      sxs += kx[k];
      sys += ky[k];
    }
    const float rx = 1.0f / sxs, ry = 1.0f / sys;
#pragma unroll
    for (int k = 0; k < KS; ++k) { kx[k] *= rx; ky[k] *= ry; }
  }

  // ---- branch-free banded-Toeplitz weight operands ----
  // vertical A chunk t:   Wv[m=idx][j=4t+2h+s]  = ky[j - idx]
  // horizontal B chunk t: Wh[r=4t+2h+s][n=idx]  = kx[r - idx]
  // same band index d per (t, slot) -> shared v_cmp chain feeds both selects
  v2f wva[KS];  // vertical-pass A operands
  v2f whb[KS];  // horizontal-pass B operands
  {
    const int dbase = 2 * half - idx;
#pragma unroll
    for (int t = 0; t < KS; ++t) {
#pragma unroll
      for (int s = 0; s < 2; ++s) {
        const int d = 4 * t + s + dbase;
        float fy = 0.0f, fx = 0.0f;
#pragma unroll
        for (int q = 0; q < KS; ++q) {
          const bool m = (d == q);
          fy = m ? ky[q] : fy;
          fx = m ? kx[q] : fx;
        }
        wva[t][s] = fy;
        whb[t][s] = fx;
      }
    }
  }

  wave_async_sync();  // patch resident in LDS

  const int ks0 = half * 2;

  // ---- vertical pass: V(16x20) = Wv(16x20) x P(20x{0..15 | 4..19}) ----
  // preload all B operands, then issue 10 back-to-back WMMAs
  v2f bmv[2][KS];
#pragma unroll
  for (int s = 0; s < 2; ++s) {
    const int c0 = s * 4;
#pragma unroll
    for (int t = 0; t < KS; ++t) {
      const int r0 = 4 * t + ks0;
      bmv[s][t][0] = sP[wave][r0 * PSTRIDE + c0 + idx];
      bmv[s][t][1] = sP[wave][(r0 + 1) * PSTRIDE + c0 + idx];
    }
  }
  v8f vacc[2];
#pragma unroll
  for (int s = 0; s < 2; ++s) {
    v8f acc = {};
#pragma unroll
    for (int t = 0; t < KS; ++t) {
      acc = __builtin_amdgcn_wmma_f32_16x16x4_f32(
          false, wva[t], false, bmv[s][t], (short)0, acc, false, false);
    }
    vacc[s] = acc;
  }

  // ---- park V(16x20) in LDS (D layout -> row major) ----
  {
    const int m0 = 8 * half;  // D layout: VGPR v -> row v + 8*half, col idx
#pragma unroll
    for (int v = 0; v < 8; ++v)
      sV[wave][(m0 + v) * VSTRIDE + idx] = vacc[0][v];
    if (idx >= 12) {  // block-1 col n -> c = 4 + n; keep c in [16,20)
#pragma unroll
      for (int v = 0; v < 8; ++v)
        sV[wave][(m0 + v) * VSTRIDE + 4 + idx] = vacc[1][v];
    }
  }
  wave_ds_sync();

  // ---- horizontal pass: out(16x16) = V(16x20) x Wh(20x16) ----
  // A operands are 8B-aligned pairs (VSTRIDE even, r0 even) -> ds_load_b64
  v2f amh[KS];
#pragma unroll
  for (int t = 0; t < KS; ++t) {
    const int r0 = 4 * t + ks0;
    amh[t] = *(const v2f*)&sV[wave][idx * VSTRIDE + r0];
  }
  v8f oacc = {};
#pragma unroll
  for (int t = 0; t < KS; ++t) {
    oacc = __builtin_amdgcn_wmma_f32_16x16x4_f32(
        false, amh[t], false, whb[t], (short)0, oacc, false, false);
  }

  // ---- store tile: per VGPR, lanes cover rows (v, v+8) x 16 contiguous cols
  float* op = out + planeOff + (size_t)y0 * IMG_W + x0;
  const int m0 = 8 * half;
#pragma unroll
  for (int v = 0; v < 8; ++v)
    op[(m0 + v) * IMG_W + idx] = oacc[v];
}

extern "C" void kernel_launch(void* const* d_in, const int* in_sizes, int n_in,
                              void* d_out, int out_size, void* d_ws, size_t ws_size,
                              hipStream_t stream) {
  const float* imgs = (const float*)d_in[0];   // [64,3,512,512] f32
  const float* xf   = (const float*)d_in[1];   // [64,2] f32
  float* out        = (float*)d_out;           // [64,3,512,512] f32
  (void)in_sizes; (void)n_in; (void)out_size; (void)d_ws; (void)ws_size;

  const int tiles  = BATCH * CH * TILES_PER_PLANE;   // 196608
  const int blocks = tiles / WAVES_PER_BLOCK;        // 24576
  gauss5x5_wmma<<<blocks, 256, 0, stream>>>(imgs, xf, out);
}